// Pbmm_26448408609097
// MI455X (gfx1250) — compile-verified
//
#include <hip/hip_runtime.h>
#include <math.h>

typedef __attribute__((ext_vector_type(2))) float v2f;
typedef __attribute__((ext_vector_type(8))) float v8f;
typedef __attribute__((ext_vector_type(4))) int   v4i;

#define WPB 8        // waves per block (phase 1)
#define TILE 256     // elements per row-tile scanned per wave iteration

// gfx1250 async global->LDS path (ASYNCcnt), guarded so compile never breaks.
#if defined(__has_builtin)
#  if __has_builtin(__builtin_amdgcn_global_load_async_to_lds_b128) && \
      __has_builtin(__builtin_amdgcn_s_wait_asynccnt)
#    define HAVE_ASYNC_LDS 1
#  endif
#endif
#ifndef HAVE_ASYNC_LDS
#  define HAVE_ASYNC_LDS 0
#endif
#define GAS __attribute__((address_space(1)))
#define LAS __attribute__((address_space(3)))

__device__ __forceinline__ int imax(int a, int b) { return a > b ? a : b; }
__device__ __forceinline__ int imin(int a, int b) { return a < b ? a : b; }

// ---------------------------------------------------------------------------
// Phase 1: per-row inclusive prefix sums via V_WMMA_F32_16X16X4_F32.
// Row tile of 256 elems viewed as 16x16 matrix M[r][c] = row[base + r*16 + c].
//   S = M x U           (U[k][j] = k<=j : per-chunk inclusive scans)
//   S += Lstrict x Trep (Trep[k][j] = chunk_total[k] : cross-chunk carries)
// Then add the running row carry and store; P[i][j] = sum(in[i][0..j]).
// ---------------------------------------------------------------------------
__global__ __launch_bounds__(256) void pbmm_rowscan_wmma(
    const float* __restrict__ in, float* __restrict__ P, int n) {
  __shared__ float stage[WPB * TILE];
  __shared__ float ltot[WPB * 16];

  const int tid  = threadIdx.x;
  const int w    = tid >> 5;
  const int lane = tid & 31;
  const int m    = lane & 15;              // A row / B,C,D column
  const int koff = (lane < 16) ? 0 : 2;    // K sub-offset per half-wave
  const int hi   = (lane < 16) ? 0 : 8;    // C/D row-half offset
  const int row  = blockIdx.x * WPB + w;

  const float* rp = in + (size_t)row * n;
  float*       wp = P  + (size_t)row * n;

  float rowCarry = 0.0f;
  const int ntiles = n / TILE;

  for (int t = 0; t < ntiles; ++t) {
    const int base = t * TILE;
    __syncthreads();
    if (t + 1 < ntiles)  // gfx1250 global_prefetch_b8 of next tile
      __builtin_prefetch(rp + base + TILE + lane, 0, 0);

#if HAVE_ASYNC_LDS
    // global_load_async_to_lds_b128: 16B/lane DMA'd straight into LDS,
    // tracked by ASYNCcnt (no VGPR round-trip).
#pragma unroll
    for (int j = 0; j < 2; ++j) {
      const int e = j * 128 + lane * 4;    // element offset within tile
      __builtin_amdgcn_global_load_async_to_lds_b128(
          (GAS v4i*)(rp + base + e),
          (LAS v4i*)&stage[w * TILE + e], 0, 0);
    }
    __builtin_amdgcn_s_wait_asynccnt(0);
#else
#pragma unroll
    for (int j = 0; j < 8; ++j)
      stage[w * TILE + j * 32 + lane] = rp[base + j * 32 + lane];
#endif
    __syncthreads();

    // S = M x U  (4 accumulating K=4 WMMAs)
    v8f S = {};
#pragma unroll
    for (int q = 0; q < 4; ++q) {
      const int k0 = 4 * q + koff;
      v2f a, b;
      a.x = stage[w * TILE + m * 16 + k0];
      a.y = stage[w * TILE + m * 16 + k0 + 1];
      b.x = (k0     <= m) ? 1.0f : 0.0f;   // U upper-triangular (incl diag)
      b.y = (k0 + 1 <= m) ? 1.0f : 0.0f;
      S = __builtin_amdgcn_wmma_f32_16x16x4_f32(false, a, false, b,
                                                (short)0, S, false, false);
    }

    // chunk totals = column 15 of S -> LDS (lanes 15 and 31 hold col 15)
    union { v8f v; float f[8]; } u;
    u.v = S;
    if (m == 15) {
#pragma unroll
      for (int v = 0; v < 8; ++v) ltot[w * 16 + hi + v] = u.f[v];
    }
    __syncthreads();

    // S += Lstrict x Trep  (cross-chunk exclusive-scan carries)
#pragma unroll
    for (int q = 0; q < 4; ++q) {
      const int k0 = 4 * q + koff;
      v2f a, b;
      a.x = (k0     < m) ? 1.0f : 0.0f;    // strictly lower triangular
      a.y = (k0 + 1 < m) ? 1.0f : 0.0f;
      b.x = ltot[w * 16 + k0];             // broadcast chunk totals
      b.y = ltot[w * 16 + k0 + 1];
      S = __builtin_amdgcn_wmma_f32_16x16x4_f32(false, a, false, b,
                                                (short)0, S, false, false);
    }

    u.v = S;
#pragma unroll
    for (int v = 0; v < 8; ++v)
      wp[base + (hi + v) * 16 + m] = u.f[v] + rowCarry;

    float tt = 0.0f;                       // advance row carry (uniform)
#pragma unroll
    for (int k = 0; k < 16; ++k) tt += ltot[w * 16 + k];
    rowCarry += tt;
  }
}

// ---------------------------------------------------------------------------
// Phase 2: sequential boundary scan (single wave32).
// ---------------------------------------------------------------------------
__device__ __forceinline__ float seg_sum(const float* __restrict__ row,
                                         int j0, int j1, int lane) {
  float acc = 0.0f;
  for (int j = j0 + lane; j < j1; j += 32) acc += row[j];
#pragma unroll
  for (int off = 16; off > 0; off >>= 1) acc += __shfl_xor(acc, off, 32);
  return acc;
}

__global__ __launch_bounds__(32) void pbmm_scan(
    const float* __restrict__ in, const float* __restrict__ P, int use_csum,
    const float* __restrict__ thp, const float* __restrict__ cbtp,
    const int* __restrict__ oimp, int* __restrict__ dout, int n) {
  __shared__ float diag[8192];             // csum[ci][ci], ci < n <= 8192
  const int lane = threadIdx.x;
  const int D = 2 * n + 2;

  for (int i = lane; i < D; i += 32) dout[i] = -1;
  if (use_csum) {
    for (int i = lane; i < n; i += 32)
      diag[i] = (i == 0) ? 0.0f : P[(size_t)i * n + (i - 1)];
  }
  __syncthreads();

  const float th  = *thp;
  const float cbt = *cbtp;
  const int   oim = *oimp;

  int ci = 0, li = -1, fc = 0, ap = 0, dn = 0;
  while (ci + 1 < n) {
    ci = ci + 1;
    const int cnt = ci - li - 1;
    float s;
    if (use_csum) {
      const float c0 = (li + 1 <= 0) ? 0.0f : P[(size_t)ci * n + li];
      s = diag[ci] - c0;
    } else {
      s = seg_sum(in + (size_t)ci * n, li + 1, ci, lane);
    }
    const float mean = (cnt <= 0) ? 1.0f : s / (float)imax(cnt, 1);
    const bool fail = mean < th;
    fc = fail ? fc + 1 : 0;
    ap = fail ? ap : ci;

    if (fc > oim) {
      // d.append(ap + 1)
      if (dn < D) dout[dn] = ap + 1;       // uniform all-lane store
      const int dn1 = dn + 1;
      const int len_cb  = ap - li;
      const int init_cb = li + 1;
      const int j0 = imin(imax(init_cb + 1, 0), n);
      const int j1 = imin(imax(init_cb + len_cb, 0), n);
      float segsum;
      if (use_csum) {
        const float a1 = (j1 <= 0) ? 0.0f : P[(size_t)init_cb * n + (j1 - 1)];
        const float a0 = (j0 <= 0) ? 0.0f : P[(size_t)init_cb * n + (j0 - 1)];
        segsum = a1 - a0;
      } else {
        segsum = seg_sum(in + (size_t)init_cb * n, j0, j1, lane);
      }
      const float cb_mean = segsum / (float)imax(len_cb - 1, 1);
      const float cb_back = (init_cb > 0)
          ? in[(size_t)init_cb * n + imax(init_cb - 1, 0)] : -1.0f;
      const bool adjust = (len_cb > 1) && (cb_mean < cbt);
      const bool swp  = adjust && (cb_back < cbt);
      const bool bump = adjust && !(cb_back < cbt);

      if (swp) {
        // aux = d.pop(); d.append(d[-1]+1 if len(d) else 1); d.append(aux)
        const int aux  = dout[imin(imax(dn1 - 1, 0), D - 1)];
        const int prev = (dn1 > 1) ? dout[imin(imax(dn1 - 2, 0), D - 1)] + 1
                                   : 1;
        if (dn1 - 1 < D) dout[dn1 - 1] = prev;
        if (dn1     < D) dout[dn1]     = aux;
        dn = dn1 + 1;
      } else if (bump) {
        const int idx = imax(dn1 - 2, 0);  // d[-2] += 1
        if (idx < D) { const int v = dout[idx]; dout[idx] = v + 1; }
        dn = dn1;
      } else {
        dn = dn1;
      }
      ci = ap;
      li = ap;
    }
  }
  if (dn < D) dout[dn] = ci + 1;           // final append
  dout[D] = dn + 1;                        // dn output (index 2n+2)
}

// ---------------------------------------------------------------------------
extern "C" void kernel_launch(void* const* d_in, const int* in_sizes, int n_in,
                              void* d_out, int out_size, void* d_ws,
                              size_t ws_size, hipStream_t stream) {
  const float* in   = (const float*)d_in[0];
  const float* thp  = (const float*)d_in[1];
  const float* cbtp = (const float*)d_in[2];
  const int*   oimp = (const int*)d_in[3];
  int* dout = (int*)d_out;

  // n from in_sizes[0] == n*n
  long long tot = (long long)in_sizes[0];
  int n = (int)(sqrt((double)tot) + 0.5);
  while ((long long)n * n > tot) --n;
  while ((long long)(n + 1) * (n + 1) <= tot) ++n;

  const size_t need = (size_t)n * (size_t)n * sizeof(float);
  const int use_csum =
      (ws_size >= need) && (n % TILE == 0) && (n % WPB == 0) && (n <= 8192);
  float* P = (float*)d_ws;

  if (use_csum) {
    pbmm_rowscan_wmma<<<n / WPB, 256, 0, stream>>>(in, P, n);
  }
  pbmm_scan<<<1, 32, 0, stream>>>(in, P, use_csum, thp, cbtp, oimp, dout, n);
}